// KAConvC_78056735637894
// MI455X (gfx1250) — compile-verified
//
#include <hip/hip_runtime.h>

// KA-conv on gfx1250: per-weight rational activations, contracted with
// V_WMMA_F32_16X16X4_F32 (numerator: 2 WMMAs over j=0..5; denominator: 1 WMMA
// over j=0..3), elementwise 1+|.| , rcp+NR, fma-accumulate on VALU.

typedef __attribute__((ext_vector_type(2))) float v2f;
typedef __attribute__((ext_vector_type(8))) float v8f;

#define B_  4
#define C_  32
#define OC_ 32
#define K_  3
#define H_  64
#define W_  64
#define KK_ (K_ * K_)

__global__ __launch_bounds__(256) void KAConv_wmma_f32_kernel(
    const float* __restrict__ x,       // [B, C, H, W]
    const float* __restrict__ nums,    // [OC*C*K*K, 6]
    const float* __restrict__ denoms,  // [OC*C*K*K, 4]
    float* __restrict__ out)           // [B, OC, H, W]
{
    const int lane = threadIdx.x & 31;
    const int wave = blockIdx.x * 8 + (threadIdx.x >> 5);

    // 2048 waves total: [b:4][h:64][wt:4][oct:2]
    const int oct = wave & 1;
    const int wt  = (wave >> 1) & 3;
    const int h   = (wave >> 3) & 63;
    const int b   = wave >> 9;

    const int half = lane >> 4;   // selects K-pair (A/B) and M-half (C/D)
    const int l15  = lane & 15;   // A: row m (oc_local); B: col n (pixel)
    const int oc0   = oct * 16;
    const int wbase = wt * 16;

    // Coefficient row for this lane's oc: oc*C*KK + c*KK + kk
    const int rowA_base = (oc0 + l15) * (C_ * KK_);

    v8f acc = (v8f)(0.0f);
    const v8f zero = (v8f)(0.0f);

    for (int c = 0; c < C_; ++c) {
        const float* __restrict__ xc = x + ((size_t)(b * C_ + c) * H_) * W_;
        for (int kh = 0; kh < K_; ++kh) {
            const int  ih  = h + kh - 1;
            const bool hin = (unsigned)ih < (unsigned)H_;
            const int  ihc = hin ? ih : 0;
            for (int kw = 0; kw < K_; ++kw) {
                const int  kk  = kh * K_ + kw;
                const int  iw  = wbase + l15 + kw - 1;
                const bool win = (unsigned)iw < (unsigned)W_;
                const int  iwc = win ? iw : 0;

                // Tap value for pixel n = lane&15 (clamped load, zero pad)
                float xv = xc[ihc * W_ + iwc];
                xv = (hin && win) ? xv : 0.0f;

                // Powers of x
                const float x2 = xv * xv;
                const float x3 = x2 * xv;
                const float x4 = x2 * x2;
                const float x5 = x4 * xv;

                // B operands, 4x16: lane n = l15, V0/V1 = K-pair by half.
                // WMMA1 rows: x^0..x^3 ; WMMA2 rows: x^4,x^5,0,0 ; den: x^1..x^4
                v2f bnum0 = { half ? x2   : 1.0f, half ? x3   : xv };
                v2f bnum1 = { half ? 0.0f : x4,   half ? 0.0f : x5 };
                v2f bden  = { half ? x3   : xv,   half ? x4   : x2 };

                // A operands: one contiguous float2 per lane from coeff rows.
                const int rowA = rowA_base + c * KK_ + kk;
                const v2f* __restrict__ ap = (const v2f*)(nums + (size_t)rowA * 6);
                v2f a01 = ap[half];        // j = {0,1} or {2,3}
                v2f a45 = ap[2];           // j = {4,5}; zero on K=6,7 half
                a45 = half ? (v2f){0.0f, 0.0f} : a45;
                const v2f* __restrict__ bp = (const v2f*)(denoms + (size_t)rowA * 4);
                v2f bcf = bp[half];        // b_{j} for j={1,2} or {3,4}

                // D = A x B (+C): numerator split across K, denominator in one.
                v8f num = __builtin_amdgcn_wmma_f32_16x16x4_f32(
                    false, a01, false, bnum0, (short)0, zero, false, false);
                num = __builtin_amdgcn_wmma_f32_16x16x4_f32(
                    false, a45, false, bnum1, (short)0, num, false, false);
                v8f den = __builtin_amdgcn_wmma_f32_16x16x4_f32(
                    false, bcf, false, bden, (short)0, zero, false, false);

                // r = num / (1 + |den|), accumulated per (oc,pixel) element.
#pragma unroll
                for (int r = 0; r < 8; ++r) {
                    const float d  = 1.0f + __builtin_fabsf(den[r]);
                    float rc = __builtin_amdgcn_rcpf(d);
                    // one Newton-Raphson step: rc = rc*(2 - d*rc)
                    rc = __builtin_fmaf(__builtin_fmaf(-d, rc, 1.0f), rc, rc);
                    acc[r] = __builtin_fmaf(num[r], rc, acc[r]);
                }
            }
        }
    }

    // C/D layout: lane half -> M in {0..7}/{8..15} at VGPR r, N = l15.
    const int ocw = oc0 + half * 8;
    float* __restrict__ op =
        out + ((size_t)(b * OC_ + ocw) * H_ + h) * W_ + wbase + l15;
#pragma unroll
    for (int r = 0; r < 8; ++r) {
        op[(size_t)r * (H_ * W_)] = acc[r];
    }
}

extern "C" void kernel_launch(void* const* d_in, const int* in_sizes, int n_in,
                              void* d_out, int out_size, void* d_ws, size_t ws_size,
                              hipStream_t stream) {
    (void)in_sizes; (void)n_in; (void)out_size; (void)d_ws; (void)ws_size;
    const float* x      = (const float*)d_in[0];
    const float* nums   = (const float*)d_in[1];
    const float* denoms = (const float*)d_in[2];
    float* out          = (float*)d_out;

    // 2048 waves = 256 blocks x 8 waves (256 threads, wave32)
    dim3 grid(256), block(256);
    KAConv_wmma_f32_kernel<<<grid, block, 0, stream>>>(x, nums, denoms, out);
}